// GATLayer_61186104099750
// MI455X (gfx1250) — compile-verified
//
#include <hip/hip_runtime.h>
#include <hip/hip_bf16.h>

#define N_I  50000
#define N_T  50000
#define INF_ 128
#define H_   4
#define F_   32
#define HF_  128
#define E_   800000

typedef __attribute__((ext_vector_type(2))) float v2f;
typedef __attribute__((ext_vector_type(8))) float v8f;

// ---------------- WMMA fp32 projection: Y(M x 128) = X(M x 128) @ W(128 x 128) -----------
// One wave per 16x16 output tile. K in steps of 4 via V_WMMA_F32_16X16X4_F32.
// A 16x4 layout: lane L holds A[M = L%16][K = 2*(L/16)+j] in vgpr j (j=0..1)
// B 4x16 layout: lane L holds B[K = 2*(L/16)+j][N = L%16] in vgpr j
// C 16x16 layout: lane L vgpr j holds C[M = 8*(L/16)+j][N = L%16]
__global__ void gat_gemm_proj(const float* __restrict__ X, const float* __restrict__ W,
                              float* __restrict__ Y, int Mrows) {
    const int wavesPerBlk = blockDim.x >> 5;
    const int wave  = blockIdx.x * wavesPerBlk + (threadIdx.x >> 5);
    const int lane  = threadIdx.x & 31;
    const int nTN   = HF_ / 16;             // 8 tiles along N
    const int tM    = (wave / nTN) * 16;
    const int tN    = (wave % nTN) * 16;
    if (tM >= Mrows) return;                // wave-uniform exit (EXEC stays all-1 for WMMA)

    const int lh  = lane >> 4;              // 0/1 half selector
    const int lm  = lane & 15;

    v8f c = {};
    const float* xrow = X + (size_t)(tM + lm) * INF_;
    #pragma unroll
    for (int k0 = 0; k0 < INF_; k0 += 4) {
        const int ka = k0 + 2 * lh;
        v2f a, b;
        a.x = xrow[ka + 0];
        a.y = xrow[ka + 1];
        b.x = W[(size_t)(ka + 0) * HF_ + tN + lm];
        b.y = W[(size_t)(ka + 1) * HF_ + tN + lm];
        c = __builtin_amdgcn_wmma_f32_16x16x4_f32(false, a, false, b,
                                                  (short)0, c, false, false);
    }
    #pragma unroll
    for (int j = 0; j < 8; ++j) {
        Y[(size_t)(tM + 8 * lh + j) * HF_ + tN + lm] = c[j];
    }
}

// ---------------- per-node attention dot: out[n,h] = dot(Hn[n,h,:], avec[h,:]) -----------
__global__ void gat_node_dot(const float* __restrict__ Hn, const float* __restrict__ avec,
                             float* __restrict__ out, int N) {
    int i = blockIdx.x * blockDim.x + threadIdx.x;       // (n,h)
    if (i >= N * H_) return;
    int n = i >> 2, h = i & 3;
    const float* row = Hn + (size_t)n * HF_ + h * F_;
    const float* av  = avec + h * F_;
    float s = 0.f;
    #pragma unroll
    for (int f = 0; f < F_; ++f) s += row[f] * av[f];
    out[i] = s;
}

// ---------------- init kernels ----------------
__global__ void gat_init_out(float* __restrict__ out, const float* __restrict__ b1,
                             const float* __restrict__ b2) {
    int i = blockIdx.x * blockDim.x + threadIdx.x;
    if (i < N_T * HF_) out[i] = b1[i & (HF_ - 1)] + b2[i & (HF_ - 1)];
}

__global__ void gat_init_seg(unsigned* __restrict__ emax, float* __restrict__ denom, int n) {
    int i = blockIdx.x * blockDim.x + threadIdx.x;
    if (i < n) { emax[i] = 0x007FFFFFu; /* enc(-inf) */ denom[i] = 0.f; }
}

// ---------------- order-preserving f32 <-> u32 encoding for atomic max ----------------
__device__ __forceinline__ unsigned enc_f(float f) {
    unsigned u = __float_as_uint(f);
    return (u & 0x80000000u) ? ~u : (u | 0x80000000u);
}
__device__ __forceinline__ float dec_f(unsigned u) {
    return (u & 0x80000000u) ? __uint_as_float(u ^ 0x80000000u) : __uint_as_float(~u);
}
__device__ __forceinline__ float lrelu(float x) { return x > 0.f ? x : 0.2f * x; }

// ---------------- pass 1: segment max over edges ----------------
__global__ void gat_edge_max(const int* __restrict__ src, const int* __restrict__ dst,
                             const float* __restrict__ el, const float* __restrict__ er,
                             unsigned* __restrict__ emax) {
    int i = blockIdx.x * blockDim.x + threadIdx.x;       // (e,h)
    if (i >= E_ * H_) return;
    int e = i >> 2, h = i & 3;
    int s = src[e], d = dst[e];
    float x = lrelu(el[s * H_ + h] + er[d * H_ + h]);
    atomicMax(&emax[d * H_ + h], enc_f(x));
}

// ---------------- pass 2: exp-sum denominator ----------------
__global__ void gat_edge_denom(const int* __restrict__ src, const int* __restrict__ dst,
                               const float* __restrict__ el, const float* __restrict__ er,
                               const unsigned* __restrict__ emax, float* __restrict__ denom) {
    int i = blockIdx.x * blockDim.x + threadIdx.x;       // (e,h)
    if (i >= E_ * H_) return;
    int e = i >> 2, h = i & 3;
    int s = src[e], d = dst[e];
    float x  = lrelu(el[s * H_ + h] + er[d * H_ + h]);
    float ex = __expf(x - dec_f(emax[d * H_ + h]));
    atomicAdd(&denom[d * H_ + h], ex);
}

// ---------------- pass 3: weighted message scatter ----------------
__global__ void gat_edge_msg(const int* __restrict__ src, const int* __restrict__ dst,
                             const float* __restrict__ el, const float* __restrict__ er,
                             const unsigned* __restrict__ emax, const float* __restrict__ denom,
                             const float* __restrict__ Hs, float* __restrict__ out) {
    int i = blockIdx.x * blockDim.x + threadIdx.x;       // (e, hf)
    if (i >= E_ * HF_) return;
    int e  = i >> 7;
    int hf = i & (HF_ - 1);
    int h  = hf >> 5;
    int s = src[e], d = dst[e];
    float x     = lrelu(el[s * H_ + h] + er[d * H_ + h]);
    float alpha = __expf(x - dec_f(emax[d * H_ + h])) / denom[d * H_ + h];
    atomicAdd(&out[(size_t)d * HF_ + hf], alpha * Hs[(size_t)s * HF_ + hf]);
}

extern "C" void kernel_launch(void* const* d_in, const int* in_sizes, int n_in,
                              void* d_out, int out_size, void* d_ws, size_t ws_size,
                              hipStream_t stream) {
    const float* feat_i = (const float*)d_in[0];
    const float* feat_t = (const float*)d_in[1];
    const float* W_i2t  = (const float*)d_in[2];
    const float* al_i2t = (const float*)d_in[3];
    const float* ar_i2t = (const float*)d_in[4];
    const float* b_i2t  = (const float*)d_in[5];
    const float* W_t2t  = (const float*)d_in[6];
    const float* al_t2t = (const float*)d_in[7];
    const float* ar_t2t = (const float*)d_in[8];
    const float* b_t2t  = (const float*)d_in[9];
    const int*   src_i2t = (const int*)d_in[10];
    const int*   dst_i2t = (const int*)d_in[11];
    const int*   src_t2t = (const int*)d_in[12];
    const int*   dst_t2t = (const int*)d_in[13];
    float* out = (float*)d_out;

    // ---- workspace carving (floats) ----
    float* ws = (float*)d_ws;
    float*    hs_i2t   = ws;                    ws += (size_t)N_I * HF_;   // feat_i @ W_i2t
    float*    hd_i2t   = ws;                    ws += (size_t)N_T * HF_;   // feat_t @ W_i2t
    float*    h_t2t    = ws;                    ws += (size_t)N_T * HF_;   // feat_t @ W_t2t
    float*    el_i2t   = ws;                    ws += (size_t)N_I * H_;
    float*    er_i2t   = ws;                    ws += (size_t)N_T * H_;
    float*    el_t2t   = ws;                    ws += (size_t)N_T * H_;
    float*    er_t2t   = ws;                    ws += (size_t)N_T * H_;
    unsigned* emax_i2t = (unsigned*)ws;         ws += (size_t)N_T * H_;
    float*    den_i2t  = ws;                    ws += (size_t)N_T * H_;
    unsigned* emax_t2t = (unsigned*)ws;         ws += (size_t)N_T * H_;
    float*    den_t2t  = ws;                    ws += (size_t)N_T * H_;

    const int BLK = 256;

    // ---- init output with summed biases; init segment buffers ----
    gat_init_out<<<(N_T * HF_ + BLK - 1) / BLK, BLK, 0, stream>>>(out, b_i2t, b_t2t);
    gat_init_seg<<<(N_T * H_ + BLK - 1) / BLK, BLK, 0, stream>>>(emax_i2t, den_i2t, N_T * H_);
    gat_init_seg<<<(N_T * H_ + BLK - 1) / BLK, BLK, 0, stream>>>(emax_t2t, den_t2t, N_T * H_);

    // ---- WMMA projections: 8 waves/block, 1 tile/wave ----
    {
        int tiles_i = (N_I / 16) * (HF_ / 16);
        int tiles_t = (N_T / 16) * (HF_ / 16);
        gat_gemm_proj<<<(tiles_i + 7) / 8, BLK, 0, stream>>>(feat_i, W_i2t, hs_i2t, N_I);
        gat_gemm_proj<<<(tiles_t + 7) / 8, BLK, 0, stream>>>(feat_t, W_i2t, hd_i2t, N_T);
        gat_gemm_proj<<<(tiles_t + 7) / 8, BLK, 0, stream>>>(feat_t, W_t2t, h_t2t, N_T);
    }

    // ---- per-node attention logits ----
    gat_node_dot<<<(N_I * H_ + BLK - 1) / BLK, BLK, 0, stream>>>(hs_i2t, al_i2t, el_i2t, N_I);
    gat_node_dot<<<(N_T * H_ + BLK - 1) / BLK, BLK, 0, stream>>>(hd_i2t, ar_i2t, er_i2t, N_T);
    gat_node_dot<<<(N_T * H_ + BLK - 1) / BLK, BLK, 0, stream>>>(h_t2t, al_t2t, el_t2t, N_T);
    gat_node_dot<<<(N_T * H_ + BLK - 1) / BLK, BLK, 0, stream>>>(h_t2t, ar_t2t, er_t2t, N_T);

    // ---- edge softmax + aggregation, relation i2t ----
    int gEH = (E_ * H_ + BLK - 1) / BLK;
    int gEF = (E_ * HF_ + BLK - 1) / BLK;
    gat_edge_max  <<<gEH, BLK, 0, stream>>>(src_i2t, dst_i2t, el_i2t, er_i2t, emax_i2t);
    gat_edge_denom<<<gEH, BLK, 0, stream>>>(src_i2t, dst_i2t, el_i2t, er_i2t, emax_i2t, den_i2t);
    gat_edge_msg  <<<gEF, BLK, 0, stream>>>(src_i2t, dst_i2t, el_i2t, er_i2t, emax_i2t, den_i2t,
                                            hs_i2t, out);

    // ---- edge softmax + aggregation, relation t2t ----
    gat_edge_max  <<<gEH, BLK, 0, stream>>>(src_t2t, dst_t2t, el_t2t, er_t2t, emax_t2t);
    gat_edge_denom<<<gEH, BLK, 0, stream>>>(src_t2t, dst_t2t, el_t2t, er_t2t, emax_t2t, den_t2t);
    gat_edge_msg  <<<gEF, BLK, 0, stream>>>(src_t2t, dst_t2t, el_t2t, er_t2t, emax_t2t, den_t2t,
                                            h_t2t, out);
}